// AffinityPropagate_68161130988179
// MI455X (gfx1250) — compile-verified
//
#include <hip/hip_runtime.h>
#include <stdint.h>

#define Bn 4
#define Cn 64
#define Hn 96
#define Wn 320
#define HW (Hn * Wn)            // 30720
#define PIX (Bn * HW)           // 122880
#define TPB 256
#define BLK_PER_IMG (HW / TPB)  // 120  (30720 % 256 == 0 -> block never crosses b)

typedef __attribute__((ext_vector_type(4))) unsigned int u32x4;
typedef __attribute__((ext_vector_type(8))) int i32x8;
typedef __attribute__((ext_vector_type(4))) int i32x4;

// 5x5 f32 matmul, fully unrolled: D = X * Y
__device__ __forceinline__ void mm5(float D[5][5], const float X[5][5], const float Y[5][5]) {
#pragma unroll
    for (int i = 0; i < 5; ++i) {
#pragma unroll
        for (int j = 0; j < 5; ++j) {
            float a = X[i][0] * Y[0][j];
            a = fmaf(X[i][1], Y[1][j], a);
            a = fmaf(X[i][2], Y[2][j], a);
            a = fmaf(X[i][3], Y[3][j], a);
            a = fmaf(X[i][4], Y[4][j], a);
            D[i][j] = a;
        }
    }
}

__global__ __launch_bounds__(TPB) void affprop_kernel(
    const float* __restrict__ guid,
    const float* __restrict__ fm0, const float* __restrict__ fm1,
    const float* __restrict__ fm2, const float* __restrict__ fm3,
    const float* __restrict__ fm4, float* __restrict__ out)
{
    __shared__ float lds_g[25 * TPB];   // 25.6 KB guidance tile

    const int tid = threadIdx.x;
    const int b   = blockIdx.x / BLK_PER_IMG;
    const int p0  = (blockIdx.x % BLK_PER_IMG) * TPB;   // pixel offset within HxW plane

    // ---------------- TDM: DMA the 25x256 guidance tile (row stride HW) into LDS ----------------
    {
        const uint64_t gaddr = (uint64_t)(uintptr_t)(guid + (size_t)b * 25 * HW + p0);
        const uint32_t laddr = (uint32_t)(uintptr_t)(&lds_g[0]);  // flat LDS ptr: low 32 bits = LDS offset
        if (tid < 32) {  // one wave issues the tensor DMA
            u32x4 g0;
            g0[0] = 1u;                                            // count=1 (valid), user mode, no gather
            g0[1] = laddr;                                         // lds_addr (bytes)
            g0[2] = (uint32_t)(gaddr & 0xffffffffu);               // global_addr[31:0]
            g0[3] = (uint32_t)((gaddr >> 32) & 0x1ffffffu)         // global_addr[56:32]
                  | (2u << 30);                                    // type = 2 ("image")
            i32x8 g1;
            g1[0] = (int)(2u << 16);                               // wg_mask=0, data_size=2 (4 bytes)
            g1[1] = (int)((uint32_t)(HW & 0xffff) << 16);          // tensor_dim0[15:0] @ bits[63:48]
            g1[2] = (int)((uint32_t)(HW >> 16) | (25u << 16));     // tensor_dim0[31:16] | tensor_dim1[15:0]
            g1[3] = (int)((uint32_t)TPB << 16);                    // tensor_dim1[31:16]=0 | tile_dim0=256
            g1[4] = (int)25u;                                      // tile_dim1=25, tile_dim2=0
            g1[5] = (int)HW;                                       // tensor_dim0_stride[31:0] = 30720
            g1[6] = 0;                                             // stride0[47:32]=0, stride1[15:0]=0
            g1[7] = 0;                                             // stride1[47:16]=0
            i32x4 z4 = {0, 0, 0, 0};
            i32x8 z8 = {0, 0, 0, 0, 0, 0, 0, 0};
            __builtin_amdgcn_tensor_load_to_lds(g0, g1, z4, z4, z8, 0);
            __builtin_amdgcn_s_wait_tensorcnt(0);
        }
        __syncthreads();
    }

    // ---------------- per-pixel: softmax gates, A = I + G, M = A^5 ----------------
    float A[5][5];
#pragma unroll
    for (int k = 0; k < 5; ++k) {
        float gk[5];
#pragma unroll
        for (int j = 0; j < 5; ++j) gk[j] = lds_g[(k * 5 + j) * TPB + tid];
        float m = gk[0];
#pragma unroll
        for (int j = 1; j < 5; ++j) m = fmaxf(m, gk[j]);
        float e[5], s = 0.f;
#pragma unroll
        for (int j = 0; j < 5; ++j) { e[j] = __expf(gk[j] - m); s += e[j]; }
        const float inv = 1.0f / s;
#pragma unroll
        for (int j = 0; j < 5; ++j) A[k][j] = e[j] * inv;
        A[k][k] += 1.0f;   // I + G
    }
    float A2[5][5], A4[5][5], M[5][5];
    mm5(A2, A, A);
    mm5(A4, A2, A2);
    mm5(M, A4, A);         // (I+G)^5

    // ---------------- apply M to the 5x64 feature matrix, single pass ----------------
    const size_t pb = (size_t)b * Cn * HW + (size_t)p0 + tid;   // [B,C,H,W] base at c=0
    const float* __restrict__ f0 = fm0 + pb;
    const float* __restrict__ f1 = fm1 + pb;
    const float* __restrict__ f2 = fm2 + pb;
    const float* __restrict__ f3 = fm3 + pb;
    const float* __restrict__ f4 = fm4 + pb;
    float* __restrict__ ob = out + pb;                          // + k*Bn*Cn*HW + c*HW

#pragma unroll 4
    for (int c = 0; c < Cn; ++c) {
        const size_t off = (size_t)c * HW;
        if (c + 8 < Cn) {   // speculative prefetch of the strided channel walk
            __builtin_prefetch(f0 + off + 8 * (size_t)HW, 0, 0);
            __builtin_prefetch(f2 + off + 8 * (size_t)HW, 0, 0);
            __builtin_prefetch(f4 + off + 8 * (size_t)HW, 0, 0);
        }
        float f[5];
        f[0] = f0[off]; f[1] = f1[off]; f[2] = f2[off]; f[3] = f3[off]; f[4] = f4[off];
#pragma unroll
        for (int k = 0; k < 5; ++k) {
            float acc = M[k][0] * f[0];
            acc = fmaf(M[k][1], f[1], acc);
            acc = fmaf(M[k][2], f[2], acc);
            acc = fmaf(M[k][3], f[3], acc);
            acc = fmaf(M[k][4], f[4], acc);
            ob[(size_t)k * (Bn * Cn * HW) + off] = acc;
        }
    }
}

extern "C" void kernel_launch(void* const* d_in, const int* in_sizes, int n_in,
                              void* d_out, int out_size, void* d_ws, size_t ws_size,
                              hipStream_t stream) {
    (void)in_sizes; (void)n_in; (void)out_size; (void)d_ws; (void)ws_size;
    const float* guid = (const float*)d_in[0];
    const float* fm0  = (const float*)d_in[1];
    const float* fm1  = (const float*)d_in[2];
    const float* fm2  = (const float*)d_in[3];
    const float* fm3  = (const float*)d_in[4];
    const float* fm4  = (const float*)d_in[5];
    float* out = (float*)d_out;

    const int blocks = PIX / TPB;   // 480
    affprop_kernel<<<blocks, TPB, 0, stream>>>(guid, fm0, fm1, fm2, fm3, fm4, out);
}